// GraphTransformer_78769700209215
// MI455X (gfx1250) — compile-verified
//
#include <hip/hip_runtime.h>
#include <hip/hip_bf16.h>

typedef __attribute__((ext_vector_type(2))) float v2f;
typedef __attribute__((ext_vector_type(8))) float v8f;

#define N_NODES 50000
#define N_EDGES 600000
#define F_IN    128
#define EMB     128
#define OUT_F   64

// ---------------------------------------------------------------------------
// Degree / normalization kernels
// ---------------------------------------------------------------------------
__global__ void k_init_deg(float* __restrict__ deg) {
    int i = blockIdx.x * blockDim.x + threadIdx.x;
    if (i < N_NODES) deg[i] = 1.0f;  // self-loop contributes 1
}

__global__ void k_deg_edges(const int* __restrict__ dst, float* __restrict__ deg) {
    int e = blockIdx.x * blockDim.x + threadIdx.x;
    if (e < N_EDGES) atomicAdd(&deg[dst[e]], 1.0f);
}

__global__ void k_dinv(float* __restrict__ deg) {
    int i = blockIdx.x * blockDim.x + threadIdx.x;
    if (i < N_NODES) deg[i] = rsqrtf(deg[i]);  // deg >= 1 always
}

// ---------------------------------------------------------------------------
// f32 WMMA GEMM:  C[M x NCOL] = A[M x 128] * W[128 x NCOL] (+bias, relu)
// One wave computes one 16x16 tile using V_WMMA_F32_16X16X4_F32, K unrolled.
// A fragment (16x4 f32): lanes 0-15 hold K=0(v0),K=1(v1); lanes 16-31 K=2,K=3.
// B fragment (4x16 f32): one row striped across lanes per VGPR (mirror of A).
// C/D (16x16 f32, 8 VGPRs): vgpr g -> row g (lanes 0-15) / row g+8 (lanes 16-31).
// ---------------------------------------------------------------------------
template <int NCOL, bool BIAS, bool RELU>
__global__ __launch_bounds__(32 * (NCOL / 16))
void k_gemm_wmma(const float* __restrict__ A, const float* __restrict__ W,
                 const float* __restrict__ bias, float* __restrict__ C) {
    constexpr int K = 128;
    const int lane = threadIdx.x & 31;
    const int wave = threadIdx.x >> 5;     // tile_n within block
    const int half = lane >> 4;            // 0: K pair {0,1}, 1: K pair {2,3}
    const int l15  = lane & 15;
    const int row0 = blockIdx.x * 16;
    const int col  = wave * 16 + l15;
    const int arow = row0 + l15;

    const float* ap = A + (size_t)arow * K + 2 * half;
    const float* wp = W + (size_t)(2 * half) * NCOL + col;

    v8f acc = {};
#pragma unroll
    for (int kb = 0; kb < K; kb += 4) {
        v2f a, b;
        a.x = ap[0];
        a.y = ap[1];
        b.x = wp[0];
        b.y = wp[NCOL];
        acc = __builtin_amdgcn_wmma_f32_16x16x4_f32(
            /*neg_a=*/false, a, /*neg_b=*/false, b,
            /*c_mod=*/(short)0, acc, /*reuse_a=*/false, /*reuse_b=*/false);
        ap += 4;
        wp += 4 * NCOL;
    }

    float bv = 0.0f;
    if (BIAS) bv = bias[col];
    (void)bias;
#pragma unroll
    for (int g = 0; g < 8; ++g) {
        int r = row0 + g + half * 8;
        float v = acc[g] + bv;
        if (RELU) v = fmaxf(v, 0.0f);
        C[(size_t)r * NCOL + col] = v;
    }
}

// ---------------------------------------------------------------------------
// agg = h * dinv^2  (self-loop term, also initializes agg)
// ---------------------------------------------------------------------------
__global__ void k_self_loop(const float* __restrict__ h, const float* __restrict__ dinv,
                            float* __restrict__ agg) {
    int idx = blockIdx.x * blockDim.x + threadIdx.x;
    if (idx < N_NODES * EMB) {
        int i = idx >> 7;  // /128
        float d = dinv[i];
        agg[idx] = h[idx] * d * d;
    }
}

// ---------------------------------------------------------------------------
// Edge scatter: one wave per edge. 32 lanes x float4 = 128 features.
// agg[dst] += h[src] * dinv[src]*dinv[dst]   via global_atomic_add_f32.
// ---------------------------------------------------------------------------
__global__ void k_edge_msg(const int* __restrict__ src, const int* __restrict__ dst,
                           const float* __restrict__ dinv,
                           const float* __restrict__ h, float* __restrict__ agg) {
    int e = blockIdx.x * (blockDim.x >> 5) + (threadIdx.x >> 5);
    if (e >= N_EDGES) return;
    int lane = threadIdx.x & 31;
    int s = src[e];
    int d = dst[e];
    float c = dinv[s] * dinv[d];
    const float4* hs = (const float4*)(h + (size_t)s * EMB);
    float4 v = hs[lane];
    float* ad = agg + (size_t)d * EMB + lane * 4;
    atomicAdd(ad + 0, v.x * c);
    atomicAdd(ad + 1, v.y * c);
    atomicAdd(ad + 2, v.z * c);
    atomicAdd(ad + 3, v.w * c);
}

// ---------------------------------------------------------------------------
// agg = relu(agg + b_gcn)
// ---------------------------------------------------------------------------
__global__ void k_bias_relu(float* __restrict__ agg, const float* __restrict__ b) {
    int idx = blockIdx.x * blockDim.x + threadIdx.x;
    if (idx < N_NODES * EMB) {
        agg[idx] = fmaxf(agg[idx] + b[idx & (EMB - 1)], 0.0f);
    }
}

// ---------------------------------------------------------------------------
// Launch
// ---------------------------------------------------------------------------
extern "C" void kernel_launch(void* const* d_in, const int* in_sizes, int n_in,
                              void* d_out, int out_size, void* d_ws, size_t ws_size,
                              hipStream_t stream) {
    (void)in_sizes; (void)n_in; (void)out_size; (void)ws_size;

    const float* x     = (const float*)d_in[0];
    const int*   ei    = (const int*)d_in[1];   // [2, E] int32 per harness convention
    const int*   srcI  = ei;
    const int*   dstI  = ei + N_EDGES;
    const float* w_gcn = (const float*)d_in[2];
    const float* b_gcn = (const float*)d_in[3];
    const float* w_lin = (const float*)d_in[4];
    const float* b_lin = (const float*)d_in[5];
    const float* w_fc  = (const float*)d_in[6];
    const float* b_fc  = (const float*)d_in[7];
    float*       out   = (float*)d_out;

    // Workspace layout (all 256B aligned):
    //   dinv : N floats              @ 0
    //   hbuf : N*128 floats          @ 204800
    //   agg  : N*128 floats          @ 204800 + 25600000
    char*  ws   = (char*)d_ws;
    float* dinv = (float*)ws;
    float* hbuf = (float*)(ws + 204800);
    float* agg  = (float*)(ws + 204800 + (size_t)N_NODES * EMB * sizeof(float));

    const int TPB = 256;

    // 1) degrees -> dinv
    k_init_deg<<<(N_NODES + TPB - 1) / TPB, TPB, 0, stream>>>(dinv);
    k_deg_edges<<<(N_EDGES + TPB - 1) / TPB, TPB, 0, stream>>>(dstI, dinv);
    k_dinv<<<(N_NODES + TPB - 1) / TPB, TPB, 0, stream>>>(dinv);

    // 2) h = x @ w_gcn   (50000/16 = 3125 row tiles, 8 col tiles per block)
    k_gemm_wmma<EMB, false, false><<<N_NODES / 16, 256, 0, stream>>>(x, w_gcn, nullptr, hbuf);

    // 3) agg = h * dinv^2 ; then edge scatter
    k_self_loop<<<(N_NODES * EMB) / TPB, TPB, 0, stream>>>(hbuf, dinv, agg);
    k_edge_msg<<<N_EDGES / 8, 256, 0, stream>>>(srcI, dstI, dinv, hbuf, agg);

    // 4) agg = relu(agg + b_gcn)
    k_bias_relu<<<(N_NODES * EMB) / TPB, TPB, 0, stream>>>(agg, b_gcn);

    // 5) h2 = relu(agg @ w_lin + b_lin)   (reuse hbuf)
    k_gemm_wmma<EMB, true, true><<<N_NODES / 16, 256, 0, stream>>>(agg, w_lin, b_lin, hbuf);

    // 6) out = h2 @ w_fc + b_fc
    k_gemm_wmma<OUT_F, true, false><<<N_NODES / 16, 128, 0, stream>>>(hbuf, w_fc, b_fc, out);
}